// OneHotDenseEncoderTorch_65300682768585
// MI455X (gfx1250) — compile-verified
//
#include <hip/hip_runtime.h>

#define OUT_DIM 64
#define DEPTH   257
#define OOV     256
#define LUT_N   512
#define ROWF    64   // floats per LDS row of W^T (== OUT_DIM, bank-conflict-free for b128 reads)

typedef float v4f __attribute__((ext_vector_type(4)));

__global__ __launch_bounds__(256)
void OneHotDenseEncoder_gather_kernel(const float* __restrict__ raw_ids,
                                      const float* __restrict__ W,
                                      const int*   __restrict__ lut,
                                      float*       __restrict__ out,
                                      int n_ids)
{
    // W^T staged in LDS: Wt[d][o] = W[o][d];  257*64*4 = 65,792 B  (+2 KB LUT) << 320 KB/WGP
    __shared__ __align__(16) float Wt[DEPTH * ROWF];
    __shared__ int lutS[LUT_N];

    const int t = threadIdx.x;   // 256 threads = 8 wave32

    // ---- stage LUT (plain ds stores) ----
    lutS[t]       = lut[t];
    lutS[t + 256] = lut[t + 256];

    // ---- stage W^T via gfx1250 async global->LDS DMA (transpose on the fly) ----
    // Per-lane independent global source + LDS destination; tracked by ASYNCcnt.
    unsigned wtBase = (unsigned)(unsigned long long)(const void*)Wt;  // low 32b of flat ptr = LDS offset
    for (int idx = t; idx < OUT_DIM * DEPTH; idx += 256) {
        int o = idx / DEPTH;              // row of W   (output dim)
        int d = idx - o * DEPTH;          // col of W   (depth/bucket)
        unsigned ldsAddr = wtBase + (unsigned)(d * (ROWF * 4) + o * 4);
        unsigned long long gaddr = (unsigned long long)(const void*)(W + idx);
        asm volatile("global_load_async_to_lds_b32 %0, %1, off"
                     :: "v"(ldsAddr), "v"(gaddr)
                     : "memory");
    }
    asm volatile("s_wait_asynccnt 0" ::: "memory");
    __syncthreads();

    // ---- streaming gather ----
    // One float4 (16 B) of one output row per lane: 16 lanes share an id,
    // each wave32 stores 512 contiguous bytes per iteration.
    const int total  = n_ids * 16;                       // float4 elements (2^25, fits int)
    const int stride = (int)(gridDim.x * blockDim.x);
    for (int e = (int)(blockIdx.x * blockDim.x) + t; e < total; e += stride) {
        int id = e >> 4;
        int c4 = e & 15;

        // prefetch next iteration's id dword (speculative; OOB is silently dropped)
        __builtin_prefetch(&raw_ids[(unsigned)(e + stride) >> 4], 0, 0);

        float r  = raw_ids[id];
        int  iid = __float2int_rn(r);
        int  b   = ((unsigned)iid < (unsigned)LUT_N) ? lutS[iid] : OOV;

        v4f v = *(const v4f*)&Wt[b * ROWF + c4 * 4];     // ds_load_b128, conflict-free
        __builtin_nontemporal_store(v, (v4f*)(out) + e); // global_store_b128 th:NT
    }
}

extern "C" void kernel_launch(void* const* d_in, const int* in_sizes, int n_in,
                              void* d_out, int out_size, void* d_ws, size_t ws_size,
                              hipStream_t stream) {
    const float* raw_ids = (const float*)d_in[0];
    const float* W       = (const float*)d_in[1];
    const int*   lut     = (const int*)d_in[2];
    float*       out     = (float*)d_out;
    const int n_ids = in_sizes[0];

    dim3 grid(1024), block(256);   // grid-stride; 128 float4-elements per thread
    hipLaunchKernelGGL(OneHotDenseEncoder_gather_kernel, grid, block, 0, stream,
                       raw_ids, W, lut, out, n_ids);
}